// GeniePathLazy_36429912605269
// MI455X (gfx1250) — compile-verified
//
#include <hip/hip_runtime.h>
#include <math.h>

// ---------------- problem constants (from reference) ----------------
#define cN0 400000
#define cN1 100000
#define cN2 25000
#define cE0 2000000
#define cE1 500000
#define cIN 128
#define cHID 16
#define cLAYERS 2
#define cSLOPE 0.2f

typedef __attribute__((ext_vector_type(2))) float v2f;
typedef __attribute__((ext_vector_type(8))) float v8f;

__device__ __forceinline__ float sigmf(float x) { return 1.0f / (1.0f + expf(-x)); }

// ---------------- fuse embed+lin1 into one [16 x 128] weight ----------------
__global__ void fuse_weights_kernel(const float* __restrict__ embW, const float* __restrict__ embB,
                                    const float* __restrict__ l1W, const float* __restrict__ l1B,
                                    float* __restrict__ Wc, float* __restrict__ bc) {
    int k = threadIdx.x; // 128 threads
    for (int o = 0; o < cHID; ++o) {
        float acc = 0.0f;
        for (int e = 0; e < 32; ++e) acc += l1W[o * 32 + e] * embW[e * cIN + k];
        Wc[o * cIN + k] = acc;
    }
    if (k < cHID) {
        float acc = l1B[k];
        for (int e = 0; e < 32; ++e) acc += l1W[k * 32 + e] * embB[e];
        bc[k] = acc;
    }
}

// ---------------- Y[M,16] = X[M,K] @ W[16,K]^T (+bias) via V_WMMA_F32_16X16X4_F32 ----------------
// One wave per 16-row tile. f32 A 16x4 frag: lanes0-15 K={0,1}, lanes16-31 K={2,3} (float2/lane).
// B 4x16 frag mirrors by N=lane&15. C/D: 8 VGPRs, m = i + (lane>=16?8:0), n = lane&15.
__global__ void proj_wmma_kernel(const float* __restrict__ X, const float* __restrict__ W,
                                 const float* __restrict__ bias, float* __restrict__ Y,
                                 int M, int K) {
    const int wid  = blockIdx.x * (blockDim.x >> 5) + (threadIdx.x >> 5);
    const int lane = threadIdx.x & 31;
    const int row0 = wid * 16;
    if (row0 >= M) return;                 // uniform per wave: EXEC stays all-ones
    const int r    = lane & 15;            // row-in-tile for A, col (n) for B/D
    const int koff = (lane >> 4) << 1;     // 0 for lanes 0-15, 2 for lanes 16-31
    int row = row0 + r;
    if (row >= M) row = M - 1;             // clamp loads on ragged tail (stores masked below)
    const float* xrow = X + (size_t)row * K;
    const float* wrow = W + (size_t)r * K;
    v8f c = {};
    for (int k0 = 0; k0 < K; k0 += 4) {
        v2f a = *(const v2f*)(xrow + k0 + koff);
        v2f b = *(const v2f*)(wrow + k0 + koff);
        c = __builtin_amdgcn_wmma_f32_16x16x4_f32(false, a, false, b, (short)0, c, false, false);
    }
    const float bn = bias ? bias[r] : 0.0f;
    const int mbase = (lane >> 4) << 3;
#pragma unroll
    for (int i = 0; i < 8; ++i) {
        int rr = row0 + mbase + i;
        if (rr < M) Y[(size_t)rr * cHID + r] = c[i] + bn;
    }
}

// ---------------- el[i] = z[i].al ; er[i] = z[i].ar (i < Ndst) ----------------
__global__ void eler_kernel(const float* __restrict__ z, const float* __restrict__ al,
                            const float* __restrict__ ar, float* __restrict__ el,
                            float* __restrict__ er, int Nsrc, int Ndst) {
    int i = blockIdx.x * blockDim.x + threadIdx.x;
    if (i >= Nsrc) return;
    const float* zi = z + (size_t)i * cHID;
    float sl = 0.0f, sr = 0.0f;
#pragma unroll
    for (int k = 0; k < cHID; ++k) { float v = zi[k]; sl += v * al[k]; sr += v * ar[k]; }
    el[i] = sl;
    if (i < Ndst) er[i] = sr;
}

// ---------------- init m=-inf, s=0, out=bias (bias survives empty segments) ----------------
__global__ void gat_init_kernel(float* __restrict__ m, float* __restrict__ s,
                                float* __restrict__ out, const float* __restrict__ b, int Ndst) {
    int gid = blockIdx.x * blockDim.x + threadIdx.x;
    if (gid >= Ndst * cHID) return;
    int i = gid >> 4, k = gid & 15;
    out[gid] = b[k];
    if (k == 0) { m[i] = __int_as_float(0xFF800000); s[i] = 0.0f; }
}

// ---------------- pass 1: e = leaky_relu(el[src]+er[dst]); segment max via int atomics ----------------
__global__ void edge_max_kernel(const int* __restrict__ src, const int* __restrict__ dst,
                                const float* __restrict__ el, const float* __restrict__ er,
                                float* __restrict__ eval, float* __restrict__ m, int E) {
    int e = blockIdx.x * blockDim.x + threadIdx.x;
    if (e >= E) return;
    int si = src[e], di = dst[e];
    float v = el[si] + er[di];
    v = (v > 0.0f) ? v : cSLOPE * v;
    eval[e] = v;
    if (v >= 0.0f) atomicMax((int*)(m + di), __float_as_int(v));
    else           atomicMin((unsigned int*)(m + di), __float_as_uint(v));
}

// ---------------- pass 2: ex = exp(e - m[dst]); s[dst] += ex ----------------
__global__ void edge_expsum_kernel(const int* __restrict__ dst, float* __restrict__ eval,
                                   const float* __restrict__ m, float* __restrict__ s, int E) {
    int e = blockIdx.x * blockDim.x + threadIdx.x;
    if (e >= E) return;
    int di = dst[e];
    float ex = expf(eval[e] - m[di]);
    eval[e] = ex;
    atomicAdd(s + di, ex);
}

// ---------------- pass 3: out[dst] += (ex/s[dst]) * z[src]  (one thread per (edge,dim)) ----------------
__global__ void edge_scatter_kernel(const int* __restrict__ src, const int* __restrict__ dst,
                                    const float* __restrict__ eval, const float* __restrict__ s,
                                    const float* __restrict__ z, float* __restrict__ out, int E) {
    int gid = blockIdx.x * blockDim.x + threadIdx.x;
    if (gid >= E * cHID) return;
    int e = gid >> 4, k = gid & 15;
    int di = dst[e];
    float alpha = eval[e] / s[di];
    atomicAdd(out + (size_t)di * cHID + k, alpha * z[(size_t)src[e] * cHID + k]);
}

// ---------------- per-node 2-layer LSTM + sigmoid head (weights staged in LDS) ----------------
__global__ void lstm_pred_kernel(const float* __restrict__ htmp, const float* __restrict__ xdin,
                                 const float* __restrict__ Wih, const float* __restrict__ Whh,
                                 const float* __restrict__ bih, const float* __restrict__ bhh,
                                 const float* __restrict__ pW, const float* __restrict__ pb,
                                 float* __restrict__ out, int N) {
    __shared__ float sWih[cLAYERS * 64 * 32];
    __shared__ float sWhh[cLAYERS * 64 * 16];
    __shared__ float sB[cLAYERS * 64];
    for (int t = threadIdx.x; t < cLAYERS * 64 * 32; t += blockDim.x) sWih[t] = Wih[t];
    for (int t = threadIdx.x; t < cLAYERS * 64 * 16; t += blockDim.x) sWhh[t] = Whh[t];
    for (int t = threadIdx.x; t < cLAYERS * 64; t += blockDim.x) sB[t] = bih[t] + bhh[t];
    __syncthreads();
    int i = blockIdx.x * blockDim.x + threadIdx.x;
    if (i >= N) return;
    float xv[16], h[16], c[16];
#pragma unroll
    for (int k = 0; k < 16; ++k) { xv[k] = xdin[(size_t)i * 16 + k]; h[k] = 0.0f; c[k] = 0.0f; }
    for (int l = 0; l < cLAYERS; ++l) {
        float inp[32];
#pragma unroll
        for (int k = 0; k < 16; ++k) {
            inp[k]      = htmp[((size_t)l * N + i) * 16 + k];
            inp[16 + k] = xv[k];
        }
        float g[64];
        const float* wih = sWih + l * 64 * 32;
        const float* whh = sWhh + l * 64 * 16;
        const float* bb  = sB + l * 64;
#pragma unroll 4
        for (int j = 0; j < 64; ++j) {
            float acc = bb[j];
#pragma unroll
            for (int k = 0; k < 32; ++k) acc += inp[k] * wih[j * 32 + k];
#pragma unroll
            for (int k = 0; k < 16; ++k) acc += h[k] * whh[j * 16 + k];
            g[j] = acc;
        }
#pragma unroll
        for (int k = 0; k < 16; ++k) {
            float ig = sigmf(g[k]);
            float fg = sigmf(g[16 + k]);
            float gg = tanhf(g[32 + k]);
            float og = sigmf(g[48 + k]);
            c[k] = fg * c[k] + ig * gg;
            h[k] = og * tanhf(c[k]);
            xv[k] = h[k];
        }
    }
    float acc = pb[0];
#pragma unroll
    for (int k = 0; k < 16; ++k) acc += h[k] * pW[k];
    out[i] = sigmf(acc);
}

// ---------------- host orchestration ----------------
extern "C" void kernel_launch(void* const* d_in, const int* in_sizes, int n_in,
                              void* d_out, int out_size, void* d_ws, size_t ws_size,
                              hipStream_t stream) {
    (void)in_sizes; (void)n_in; (void)out_size; (void)ws_size;
    const float* x        = (const float*)d_in[0];
    const float* feat_dst = (const float*)d_in[1];
    const int*   e0_src   = (const int*)d_in[2];
    const int*   e0_dst   = (const int*)d_in[3];
    const int*   e1_src   = (const int*)d_in[4];
    const int*   e1_dst   = (const int*)d_in[5];
    const float* embed_W  = (const float*)d_in[6];
    const float* embed_b  = (const float*)d_in[7];
    const float* lin1_W   = (const float*)d_in[8];
    const float* lin1_b   = (const float*)d_in[9];
    const float* gat_W    = (const float*)d_in[10];   // [2][16][16]
    const float* gat_al   = (const float*)d_in[11];   // [2][1][16]
    const float* gat_ar   = (const float*)d_in[12];
    const float* gat_b    = (const float*)d_in[13];
    const float* lstm_Wih = (const float*)d_in[14];
    const float* lstm_Whh = (const float*)d_in[15];
    const float* lstm_bih = (const float*)d_in[16];
    const float* lstm_bhh = (const float*)d_in[17];
    const float* pred_W   = (const float*)d_in[18];
    const float* pred_b   = (const float*)d_in[19];
    float* out = (float*)d_out;

    // workspace carve-out
    char* ws = (char*)d_ws;
    size_t off = 0;
    auto carve = [&](size_t bytes) -> float* {
        float* p = (float*)(ws + off);
        off += (bytes + 255) & ~(size_t)255;
        return p;
    };
    float* h0    = carve((size_t)cN0 * cHID * 4);   // proj of x
    float* z     = carve((size_t)cN0 * cHID * 4);   // GAT z (max Nsrc)
    float* el    = carve((size_t)cN0 * 4);
    float* er    = carve((size_t)cN1 * 4);
    float* mbuf  = carve((size_t)cN1 * 4);
    float* sbuf  = carve((size_t)cN1 * 4);
    float* tbuf  = carve((size_t)cN1 * cHID * 4);   // block0 output
    float* htmp  = carve((size_t)cLAYERS * cN2 * cHID * 4);
    float* xd    = carve((size_t)cN2 * cHID * 4);
    float* evals = carve((size_t)cE0 * 4);
    float* Wc    = carve((size_t)cHID * cIN * 4);
    float* bc    = carve((size_t)cHID * 4);

    const int TB = 256;
    auto projGrid = [](int M) { return (M + 127) / 128; };  // 8 waves/block * 16 rows/wave

    // fused embed+lin1, then h0 / xd projections (K=128) — the big 205MB read of x
    fuse_weights_kernel<<<1, 128, 0, stream>>>(embed_W, embed_b, lin1_W, lin1_b, Wc, bc);
    proj_wmma_kernel<<<projGrid(cN0), TB, 0, stream>>>(x, Wc, bc, h0, cN0, cIN);
    proj_wmma_kernel<<<projGrid(cN2), TB, 0, stream>>>(feat_dst, Wc, bc, xd, cN2, cIN);

    for (int l = 0; l < cLAYERS; ++l) {
        const float* Wl = gat_W + l * cHID * cHID;
        const float* al = gat_al + l * cHID;
        const float* ar = gat_ar + l * cHID;
        const float* bl = gat_b + l * cHID;

        // ---- block 0: h0 [N0] -> tbuf [N1] over e0 ----
        proj_wmma_kernel<<<projGrid(cN0), TB, 0, stream>>>(h0, Wl, nullptr, z, cN0, cHID);
        eler_kernel<<<(cN0 + TB - 1) / TB, TB, 0, stream>>>(z, al, ar, el, er, cN0, cN1);
        gat_init_kernel<<<(cN1 * cHID + TB - 1) / TB, TB, 0, stream>>>(mbuf, sbuf, tbuf, bl, cN1);
        edge_max_kernel<<<(cE0 + TB - 1) / TB, TB, 0, stream>>>(e0_src, e0_dst, el, er, evals, mbuf, cE0);
        edge_expsum_kernel<<<(cE0 + TB - 1) / TB, TB, 0, stream>>>(e0_dst, evals, mbuf, sbuf, cE0);
        edge_scatter_kernel<<<(cE0 * cHID + TB - 1) / TB, TB, 0, stream>>>(e0_src, e0_dst, evals, sbuf, z, tbuf, cE0);

        // ---- block 1: tbuf [N1] -> htmp[l] [N2] over e1 ----
        float* outl = htmp + (size_t)l * cN2 * cHID;
        proj_wmma_kernel<<<projGrid(cN1), TB, 0, stream>>>(tbuf, Wl, nullptr, z, cN1, cHID);
        eler_kernel<<<(cN1 + TB - 1) / TB, TB, 0, stream>>>(z, al, ar, el, er, cN1, cN2);
        gat_init_kernel<<<(cN2 * cHID + TB - 1) / TB, TB, 0, stream>>>(mbuf, sbuf, outl, bl, cN2);
        edge_max_kernel<<<(cE1 + TB - 1) / TB, TB, 0, stream>>>(e1_src, e1_dst, el, er, evals, mbuf, cE1);
        edge_expsum_kernel<<<(cE1 + TB - 1) / TB, TB, 0, stream>>>(e1_dst, evals, mbuf, sbuf, cE1);
        edge_scatter_kernel<<<(cE1 * cHID + TB - 1) / TB, TB, 0, stream>>>(e1_src, e1_dst, evals, sbuf, z, outl, cE1);
    }

    lstm_pred_kernel<<<(cN2 + TB - 1) / TB, TB, 0, stream>>>(htmp, xd, lstm_Wih, lstm_Whh,
                                                             lstm_bih, lstm_bhh, pred_W, pred_b,
                                                             out, cN2);
}